// ConvAttnPool_44117904065193
// MI455X (gfx1250) — compile-verified
//
#include <hip/hip_runtime.h>
#include <math.h>

typedef float v2f __attribute__((ext_vector_type(2)));
typedef float v8f __attribute__((ext_vector_type(8)));
typedef unsigned int v4u __attribute__((ext_vector_type(4)));
typedef int v4i __attribute__((ext_vector_type(4)));
typedef int v8i __attribute__((ext_vector_type(8)));

constexpr int Bc   = 16;
constexpr int Lc   = 2500;
constexpr int Ec   = 100;
constexpr int Fm   = 50;
constexpr int Yc   = 8921;
constexpr int Kc   = 10;
constexpr int PADc = 5;
constexpr int Lp   = Lc + 1;      // 2501 (even kernel, pad 5/5)
constexpr int LT   = 157;         // 16-wide l tiles
constexpr int Lpad = LT * 16;     // 2512
constexpr int FP   = 64;          // Fm padded to WMMA-K multiple
constexpr int YP   = 8960;        // Y padded to 16*8*70
constexpr int WV   = 8;           // waves (y-tiles) per block

// ---------------------------------------------------------------- pad weights
__global__ __launch_bounds__(256)
void pad_kernel(const float* __restrict__ U, const float* __restrict__ Fw,
                float* __restrict__ Upad, float* __restrict__ Fpad) {
  int idx = blockIdx.x * 256 + threadIdx.x;
  if (idx >= YP * FP) return;
  int y = idx / FP, f = idx % FP;
  bool v = (y < Yc) && (f < Fm);
  float uv = v ? U[y * Fm + f]  : 0.0f;
  float fv = v ? Fw[y * Fm + f] : 0.0f;
  Upad[idx] = uv;
  Fpad[idx] = fv;
}

// --------------------------------------- transpose conv_w (f,e,k) -> (f,k,e)
__global__ __launch_bounds__(256)
void prep_cw_kernel(const float* __restrict__ cw, float* __restrict__ cwt) {
  int idx = blockIdx.x * 256 + threadIdx.x;
  if (idx >= Fm * Kc * Ec) return;
  int f = idx / (Kc * Ec);
  int r = idx % (Kc * Ec);
  int k = r / Ec, e = r % Ec;
  cwt[idx] = cw[(f * Ec + e) * Kc + k];
}

// ------------------------------------------------------- conv1d + bias + tanh
__global__ __launch_bounds__(256)
void conv_kernel(const float* __restrict__ x, const float* __restrict__ cwt,
                 const float* __restrict__ cb, float* __restrict__ hbuf) {
  int idx = blockIdx.x * 256 + threadIdx.x;            // (b*Lpad + l)*FP + f
  int b = idx / (Lpad * FP);
  int r = idx % (Lpad * FP);
  int l = r / FP, f = r % FP;
  float acc = 0.0f;
  if (l < Lp && f < Fm) {
    acc = cb[f];
    #pragma unroll
    for (int k = 0; k < Kc; ++k) {
      int la = l + k - PADc;
      if (la >= 0 && la < Lc) {
        const float4* xr = (const float4*)(x + ((size_t)b * Lc + la) * Ec);
        const float4* wr = (const float4*)(cwt + ((size_t)f * Kc + k) * Ec);
        #pragma unroll 5
        for (int e4 = 0; e4 < Ec / 4; ++e4) {
          float4 xv = xr[e4], wv = wr[e4];
          acc = fmaf(xv.x, wv.x, acc);
          acc = fmaf(xv.y, wv.y, acc);
          acc = fmaf(xv.z, wv.z, acc);
          acc = fmaf(xv.w, wv.w, acc);
        }
      }
    }
    acc = tanhf(acc);
  }
  hbuf[idx] = acc;                                     // zeros in padded region
}

// ---------------- TDM: DMA one 16x64-float tile global->LDS, row-padded to 68
// D# pad fields: interval code 5 = 64 dwords, amount code 3 = 4 dwords ->
// reproduces the [16][68] bank-conflict-free layout in hardware.
__device__ __forceinline__ void tdm_load_tile(const float* gsrc, unsigned lds_byte_off) {
  unsigned long long ga = (unsigned long long)(uintptr_t)gsrc;
  v4u g0;
  g0.x = 1u;                                           // count=1, user D#
  g0.y = lds_byte_off;                                 // lds_addr
  g0.z = (unsigned)(ga & 0xFFFFFFFFu);                 // global_addr[31:0]
  g0.w = (unsigned)((ga >> 32) & 0x1FFFFFFu) | (2u << 30);  // addr hi, type=2
  v8i g1;
  g1[0] = (2 << 16) | (1 << 20) | (5 << 22) | (3 << 25);  // 4B elems, pad 64+4
  g1[1] = (64 & 0xFFFF) << 16;                         // tensor_dim0 = 64
  g1[2] = (16 << 16);                                  // tensor_dim1 = 16
  g1[3] = (64 << 16);                                  // tile_dim0 = 64
  g1[4] = 16;                                          // tile_dim1 = 16
  g1[5] = 64;                                          // tensor_dim0_stride = 64
  g1[6] = 0;
  g1[7] = 0;
  v4i gz = {0, 0, 0, 0};
#if defined(__clang_major__) && (__clang_major__ >= 23)
  v8i gz8 = {0, 0, 0, 0, 0, 0, 0, 0};
  __builtin_amdgcn_tensor_load_to_lds(g0, g1, gz, gz, gz8, 0);
#else
  __builtin_amdgcn_tensor_load_to_lds(g0, g1, gz, gz, 0);
#endif
}

// --------------------------------- fused attention: scores->softmax->alpha->m
__global__ __launch_bounds__(256)
void attn_kernel(const float* __restrict__ hbuf, const float* __restrict__ Upad,
                 const float* __restrict__ Fpad, const float* __restrict__ fb,
                 float* __restrict__ logits, float* __restrict__ alpha_out) {
  __shared__ float sh_h[2][16][68];       // double-buffered h tile (TDM target)
  __shared__ float sh_a[WV][16][17];      // per-wave alpha staging (D->A relay)

  const int tid  = threadIdx.x;
  const int w    = tid >> 5;
  const int lane = tid & 31;
  const int b    = blockIdx.y;
  const int y0   = (blockIdx.x * WV + w) * 16;
  const int n    = lane & 15;             // N index / A-matrix M index
  const int g    = lane >> 4;             // half-wave group
  const int koff = g * 2;                 // K offset of this half-wave

  const float* hb = hbuf + (size_t)b * Lpad * FP;
  const unsigned lds_off[2] = {
    (unsigned)(uintptr_t)&sh_h[0][0][0],
    (unsigned)(uintptr_t)&sh_h[1][0][0]
  };

  // Preload A fragments: U tile (16 y x 64 k), reused across all 157 l-tiles.
  v2f af[16];
  #pragma unroll
  for (int ks = 0; ks < 16; ++ks) {
    const float* p = Upad + (size_t)(y0 + n) * FP + ks * 4 + koff;
    af[ks].x = p[0];
    af[ks].y = p[1];
  }

  float rmax[8], rsum[8], rinv[8];
  #pragma unroll
  for (int j = 0; j < 8; ++j) { rmax[j] = -3.0e38f; rsum[j] = 0.0f; }

  // -------- pass 1: scores via WMMA + lane-local online softmax --------
  if (w == 0) tdm_load_tile(hb, lds_off[0]);
  for (int lt = 0; lt < LT; ++lt) {
    const int p = lt & 1;
    if (w == 0) {
      if (lt + 1 < LT) {
        tdm_load_tile(hb + (size_t)(lt + 1) * 16 * FP, lds_off[1 - p]);
        __builtin_amdgcn_s_wait_tensorcnt(1);
      } else {
        __builtin_amdgcn_s_wait_tensorcnt(0);
      }
    }
    __syncthreads();

    v8f d = {0.f, 0.f, 0.f, 0.f, 0.f, 0.f, 0.f, 0.f};
    #pragma unroll
    for (int ks = 0; ks < 16; ++ks) {
      v2f bf = *(const v2f*)&sh_h[p][n][ks * 4 + koff]; // B[k][n] = h[l0+n][k]
      d = __builtin_amdgcn_wmma_f32_16x16x4_f32(false, af[ks], false, bf,
                                                (short)0, d, false, false);
    }

    const bool lvalid = (lt * 16 + n) < Lp;
    #pragma unroll
    for (int j = 0; j < 8; ++j) {                       // row = j + 8*g
      float sv = lvalid ? d[j] : -3.0e38f;
      float nm = fmaxf(rmax[j], sv);
      rsum[j]  = rsum[j] * __expf(rmax[j] - nm) + __expf(sv - nm);
      rmax[j]  = nm;
    }
    __syncthreads();                    // tile buffer free for next TDM write
  }

  // merge (max,sum) pairs across the 16-lane half-group, once
  #pragma unroll
  for (int j = 0; j < 8; ++j) {
    float M = rmax[j], S = rsum[j];
    #pragma unroll
    for (int mask = 1; mask <= 8; mask <<= 1) {
      float Mo = __shfl_xor(M, mask, 32);
      float So = __shfl_xor(S, mask, 32);
      float Mn = fmaxf(M, Mo);
      S = S * __expf(M - Mn) + So * __expf(Mo - Mn);
      M = Mn;
    }
    rmax[j] = M;
    rinv[j] = 1.0f / S;
  }

  v8f macc[4];
  #pragma unroll
  for (int t = 0; t < 4; ++t)
    macc[t] = (v8f){0.f, 0.f, 0.f, 0.f, 0.f, 0.f, 0.f, 0.f};

  // -------- pass 2: alpha out (streaming NT) + m = alpha x h via WMMA --------
  if (w == 0) tdm_load_tile(hb, lds_off[0]);
  for (int lt = 0; lt < LT; ++lt) {
    const int p = lt & 1;
    if (w == 0) {
      if (lt + 1 < LT) {
        tdm_load_tile(hb + (size_t)(lt + 1) * 16 * FP, lds_off[1 - p]);
        __builtin_amdgcn_s_wait_tensorcnt(1);
      } else {
        __builtin_amdgcn_s_wait_tensorcnt(0);
      }
    }
    __syncthreads();

    v8f d = {0.f, 0.f, 0.f, 0.f, 0.f, 0.f, 0.f, 0.f};
    #pragma unroll
    for (int ks = 0; ks < 16; ++ks) {
      v2f bf = *(const v2f*)&sh_h[p][n][ks * 4 + koff];
      d = __builtin_amdgcn_wmma_f32_16x16x4_f32(false, af[ks], false, bf,
                                                (short)0, d, false, false);
    }

    const int l = lt * 16 + n;
    #pragma unroll
    for (int j = 0; j < 8; ++j) {
      int   row = j + 8 * g;
      float a   = (l < Lp) ? __expf(d[j] - rmax[j]) * rinv[j] : 0.0f;
      int   y   = y0 + row;
      if (l < Lp && y < Yc)
        __builtin_nontemporal_store(a, &alpha_out[((size_t)(b * Yc + y)) * Lp + l]);
      sh_a[w][row][n] = a;                          // relay D-layout -> A-layout
    }

    #pragma unroll
    for (int ks = 0; ks < 4; ++ks) {                // K = l within tile
      v2f aa;
      aa.x = sh_a[w][n][ks * 4 + koff];
      aa.y = sh_a[w][n][ks * 4 + koff + 1];
      #pragma unroll
      for (int t = 0; t < 4; ++t) {                 // N = 16-wide f tiles
        v2f bb;
        bb.x = sh_h[p][ks * 4 + koff][t * 16 + n];
        bb.y = sh_h[p][ks * 4 + koff + 1][t * 16 + n];
        macc[t] = __builtin_amdgcn_wmma_f32_16x16x4_f32(false, aa, false, bb,
                                                        (short)0, macc[t],
                                                        false, false);
      }
    }
    __syncthreads();
  }

  // -------- fused logits: sum_f final_w[y][f] * m[y][f] + final_b --------
  #pragma unroll
  for (int j = 0; j < 8; ++j) {
    int   row = j + 8 * g;
    int   y   = y0 + row;
    float pv  = 0.0f;
    #pragma unroll
    for (int t = 0; t < 4; ++t)
      pv = fmaf(Fpad[(size_t)y * FP + t * 16 + n], macc[t][j], pv);
    pv += __shfl_xor(pv, 1, 32);
    pv += __shfl_xor(pv, 2, 32);
    pv += __shfl_xor(pv, 4, 32);
    pv += __shfl_xor(pv, 8, 32);
    if (n == 0) {
      float bias = (y < Yc) ? fb[y] : 0.0f;
      logits[(size_t)b * YP + y] = pv + bias;
    }
  }
}

// ---------------------------------------------- log_softmax + BCE (per batch)
__global__ __launch_bounds__(256)
void head_kernel(const float* __restrict__ logits, const float* __restrict__ target,
                 float* __restrict__ proba, float* __restrict__ lpart) {
  __shared__ float red[256];
  int b = blockIdx.x, tid = threadIdx.x;
  const float* lg = logits + (size_t)b * YP;

  float m = -3.0e38f;
  for (int y = tid; y < Yc; y += 256) m = fmaxf(m, lg[y]);
  red[tid] = m; __syncthreads();
  for (int s = 128; s > 0; s >>= 1) {
    if (tid < s) red[tid] = fmaxf(red[tid], red[tid + s]);
    __syncthreads();
  }
  float gmax = red[0]; __syncthreads();

  float s = 0.0f;
  for (int y = tid; y < Yc; y += 256) s += expf(lg[y] - gmax);
  red[tid] = s; __syncthreads();
  for (int st = 128; st > 0; st >>= 1) {
    if (tid < st) red[tid] += red[tid + st];
    __syncthreads();
  }
  float lse = gmax + logf(red[0]); __syncthreads();

  float lb = 0.0f;
  for (int y = tid; y < Yc; y += 256) {
    float z = lg[y];
    proba[(size_t)b * Yc + y] = z - lse;
    float t = target[(size_t)b * Yc + y];
    lb += fmaxf(z, 0.0f) - z * t + log1pf(expf(-fabsf(z)));
  }
  red[tid] = lb; __syncthreads();
  for (int st = 128; st > 0; st >>= 1) {
    if (tid < st) red[tid] += red[tid + st];
    __syncthreads();
  }
  if (tid == 0) lpart[b] = red[0];
}

__global__ void loss_kernel(const float* __restrict__ lpart, float* __restrict__ out_loss) {
  if (threadIdx.x == 0) {
    float s = 0.0f;
    for (int i = 0; i < Bc; ++i) s += lpart[i];
    *out_loss = s / (float)(Bc * Yc);
  }
}

// ----------------------------------------------------------------------------
extern "C" void kernel_launch(void* const* d_in, const int* in_sizes, int n_in,
                              void* d_out, int out_size, void* d_ws, size_t ws_size,
                              hipStream_t stream) {
  const float* x       = (const float*)d_in[0];
  const float* target  = (const float*)d_in[1];
  const float* conv_w  = (const float*)d_in[2];
  const float* conv_b  = (const float*)d_in[3];
  const float* U_w     = (const float*)d_in[4];
  const float* final_w = (const float*)d_in[5];
  const float* final_b = (const float*)d_in[6];

  float* out   = (float*)d_out;
  float* proba = out;                               // B*Y
  float* loss  = out + (size_t)Bc * Yc;             // 1
  float* alpha = loss + 1;                          // B*Y*Lp

  float* ws     = (float*)d_ws;
  float* hbuf   = ws;                               // B*Lpad*FP
  float* Upad   = hbuf   + (size_t)Bc * Lpad * FP;  // YP*FP
  float* Fpad   = Upad   + (size_t)YP * FP;         // YP*FP
  float* logits = Fpad   + (size_t)YP * FP;         // B*YP
  float* lpart  = logits + (size_t)Bc * YP;         // B
  float* cwt    = lpart  + Bc;                      // Fm*Kc*Ec

  pad_kernel   <<<(YP * FP + 255) / 256, 256, 0, stream>>>(U_w, final_w, Upad, Fpad);
  prep_cw_kernel<<<(Fm * Kc * Ec + 255) / 256, 256, 0, stream>>>(conv_w, cwt);
  conv_kernel  <<<(Bc * Lpad * FP) / 256, 256, 0, stream>>>(x, cwt, conv_b, hbuf);
  dim3 g2(YP / (16 * WV), Bc);                      // (70, 16)
  attn_kernel  <<<g2, 256, 0, stream>>>(hbuf, Upad, Fpad, final_b, logits, alpha);
  head_kernel  <<<Bc, 256, 0, stream>>>(logits, target, proba, lpart);
  loss_kernel  <<<1, 64, 0, stream>>>(lpart, loss);
}